// QCONVbk_35880156791520
// MI455X (gfx1250) — compile-verified
//
#include <hip/hip_runtime.h>

// Quanvolution closed form:
//   RX(a)RX(w) = RX(a+w)  (product state survives both RX stages)
//   CNOT-ring Heisenberg: Z0 -> Z1*Z2*...*Z8   (wire 0 drops out)
//   <Z1..Z8> on product state = prod_{q=1..8} cos(patch_q + w_q)
// Then out[b,f,h,w] = sum_c z[b,c,h,w] broadcast over f=0..7, which is
// ones(16x4) @ Z(4x16) == V_WMMA_F32_16X16X4_F32 with K = #channels = 4.

typedef __attribute__((ext_vector_type(2))) float v2f;
typedef __attribute__((ext_vector_type(8))) float v8f;

#define BB 8
#define CC 4
#define HH 32
#define WW 32
#define FF 8

__global__ __launch_bounds__(256, 1)
void qconv_wmma_kernel(const float* __restrict__ x,
                       const float* __restrict__ wts,
                       float* __restrict__ out) {
  const int lid = threadIdx.x & 31;                       // lane in wave32
  const int gw  = (blockIdx.x * blockDim.x + threadIdx.x) >> 5;  // global wave id
  const int i0  = gw << 4;                                // first of 16 pixels
  const int b   = i0 >> 10;                               // batch (HW = 1024)
  const int hwBase = i0 & 1023;                           // tile stays in one row
  const int h   = hwBase >> 5;
  const int wb  = hwBase & 31;                            // 0 or 16

  const int n   = lid & 15;                               // pixel / B-column
  const int cp  = lid >> 4;                               // channel pair 0/1
  const int w   = wb + n;

  // circuit weights for wires 1..8 (wire 0 eliminated by the algebra);
  // uniform address -> scalarized to s_load by the compiler.
  float wt[8];
#pragma unroll
  for (int r = 0; r < 8; ++r) wt[r] = wts[r + 1];

  // Branchless 'same' padding: clamp coords to a valid address, always load,
  // then cndmask the value to 0 where the true coordinate was out of bounds.
  // -> 16 unconditional clause-able global_load_b32, no exec-mask branches.
  v2f bz;
#pragma unroll
  for (int k = 0; k < 2; ++k) {
    const int c = cp * 2 + k;
    const float* __restrict__ xc = x + ((b * CC + c) * HH * WW);
    float prod = 1.0f;
#pragma unroll
    for (int r = 1; r < 9; ++r) {                         // wires 1..8
      const int dp = r / 3 - 1;                           // patch row offset
      const int dq = r % 3 - 1;                           // patch col offset
      const int hh = h + dp;
      const int ww = w + dq;
      const int hc = min(max(hh, 0), HH - 1);
      const int wc = min(max(ww, 0), WW - 1);
      float v = xc[hc * WW + wc];                         // always in-bounds
      const bool oob = (hh != hc) | (ww != wc);
      v = oob ? 0.0f : v;                                 // zero padding
      prod *= __cosf(v + wt[r - 1]);
    }
    bz[k] = prod;
  }

  // D = ones(16x4) x Z(4x16): every row of D holds s(pixel n) at column n.
  const v2f av = {1.0f, 1.0f};
  v8f acc = {};
  acc = __builtin_amdgcn_wmma_f32_16x16x4_f32(
      /*neg_a=*/false, av, /*neg_b=*/false, bz,
      /*c_mod=*/(short)0, acc, /*reuse_a=*/false, /*reuse_b=*/false);

  // acc[0]: lanes 0-15 -> (M=0, N=lane), lanes 16-31 -> (M=8, N=lane-16),
  // so every lane holds s(pixel n). Four fully-coalesced 32-lane stores
  // cover filters {0..7} x pixels {0..15}.
  const float s = acc[0];
  float* __restrict__ ob = out + (size_t)b * (FF * HH * WW) + hwBase;
#pragma unroll
  for (int f0 = 0; f0 < FF; f0 += 2) {
    ob[(f0 + cp) * (HH * WW) + n] = s;                    // 2x 64B runs / store
  }
}

extern "C" void kernel_launch(void* const* d_in, const int* in_sizes, int n_in,
                              void* d_out, int out_size, void* d_ws, size_t ws_size,
                              hipStream_t stream) {
  (void)in_sizes; (void)n_in; (void)out_size; (void)d_ws; (void)ws_size;
  const float* x   = (const float*)d_in[0];   // (8,4,32,32) f32
  const float* wts = (const float*)d_in[1];   // (1,9) f32
  float* out = (float*)d_out;                 // (8,8,32,32) f32
  // 8192 pixel sites, 16 per wave -> 512 waves -> 64 blocks of 256 (8 waves)
  qconv_wmma_kernel<<<64, 256, 0, stream>>>(x, wts, out);
}